// SimPool_45423574123184
// MI455X (gfx1250) — compile-verified
//
#include <hip/hip_runtime.h>
#include <hip/hip_bf16.h>
#include <math.h>

// ---------------------------------------------------------------------------
// SimPool single-query attention, algebraically reduced (Wk folded into q).
// B=32, H=W=64 (L=4096), C=768, heads=12, d=64, scale=1/8, eps=1e-5.
// Bandwidth-bound: 2 passes over x (805MB) ~= 35us @ 23.3TB/s.
// Logits GEMM: v_wmma_f32_16x16x4_f32; B-matrix staged via async global->LDS.
// ---------------------------------------------------------------------------

typedef __attribute__((ext_vector_type(2))) float v2f;
typedef __attribute__((ext_vector_type(8))) float v8f;

#define Bn   32
#define Ln   4096
#define Cn   768
#define Hn   12
#define Dn   64
#define NP   16          // heads padded to one WMMA N-tile
#define CHT  32          // tokens per chunk in fused pass
#define NCH  (Ln / CHT)  // 128 chunks per batch
#define PAD  772         // LDS row pitch (772 % 64 == 4 -> conflict-free cols)
#define SCALE 0.125f
#define EPS   1e-5f

// ---- workspace layout (floats) --------------------------------------------
#define WS_POOL  0                                   // [32][768]
#define WS_QS    (WS_POOL + Bn*Cn)                   // [32][768]  scaled q
#define WS_QK    (WS_QS + Bn*Cn)                     // [32][768][16] c-major
#define WS_QB    (WS_QK + Bn*Cn*NP)                  // [32][16]
#define WS_PM    (WS_QB + Bn*NP)                     // [32][128][16] chunk max
#define WS_PS    (WS_PM + Bn*NCH*NP)                 // [32][128][16] chunk sum
#define WS_PO    (WS_PS + Bn*NCH*NP)                 // [32][128][768] partial V

// ---------------------------------------------------------------------------
__global__ void zero_pool(float* pool) {
    int i = blockIdx.x * blockDim.x + threadIdx.x;
    if (i < Bn * Cn) pool[i] = 0.0f;
}

// grid (32 lchunks, 32 b), block 192: each thread owns 4 channels, sums 128 tokens
__global__ __launch_bounds__(192) void pool_kernel(const float* __restrict__ x,
                                                   float* __restrict__ pool) {
    int b = blockIdx.y, lc = blockIdx.x, tid = threadIdx.x;
    const float* xp = x + ((size_t)b * Ln + (size_t)lc * 128) * Cn + tid * 4;
    float4 acc = make_float4(0.f, 0.f, 0.f, 0.f);
    for (int l = 0; l < 128; ++l) {
        float4 v = *(const float4*)(xp + (size_t)l * Cn);
        acc.x += v.x; acc.y += v.y; acc.z += v.z; acc.w += v.w;
    }
    float* p = pool + b * Cn + tid * 4;
    atomicAdd(p + 0, acc.x); atomicAdd(p + 1, acc.y);
    atomicAdd(p + 2, acc.z); atomicAdd(p + 3, acc.w);
}

// grid (32 b), block 256: qs[b][j] = ((pool[b]/4096) @ Wq[:,j] + bq[j]) * scale
__global__ __launch_bounds__(256) void qproj_kernel(const float* __restrict__ pool,
                                                    const float* __restrict__ Wq,
                                                    const float* __restrict__ bq,
                                                    float* __restrict__ qs) {
    int b = blockIdx.x, tid = threadIdx.x;
    const float inv = 1.0f / (float)Ln;
    for (int j = tid; j < Cn; j += 256) {
        float acc = bq[j];
        for (int c = 0; c < Cn; ++c)
            acc += (pool[b * Cn + c] * inv) * Wq[(size_t)c * Cn + j];
        qs[b * Cn + j] = acc * SCALE;
    }
}

// grid (16 n, 32 b), block 256: qk[b][c][n] = sum_d qs[b,n*64+d]*Wk[c,n*64+d]
__global__ __launch_bounds__(256) void qkbuild_kernel(const float* __restrict__ qs,
                                                      const float* __restrict__ Wk,
                                                      const float* __restrict__ bk,
                                                      float* __restrict__ qk,
                                                      float* __restrict__ qb) {
    int n = blockIdx.x, b = blockIdx.y, tid = threadIdx.x;
    if (n >= Hn) {  // padded heads are zero
        for (int c = tid; c < Cn; c += 256) qk[((size_t)b * Cn + c) * NP + n] = 0.f;
        if (tid == 0) qb[b * NP + n] = 0.f;
        return;
    }
    const float* qh = qs + b * Cn + n * Dn;
    for (int c = tid; c < Cn; c += 256) {
        const float* wr = Wk + (size_t)c * Cn + n * Dn;
        float acc = 0.f;
        for (int d = 0; d < Dn; ++d) acc += qh[d] * wr[d];
        qk[((size_t)b * Cn + c) * NP + n] = acc;
    }
    if (tid == 0) {
        float acc = 0.f;
        for (int d = 0; d < Dn; ++d) acc += qh[d] * bk[n * Dn + d];
        qb[b * NP + n] = acc;
    }
}

// ---------------------------------------------------------------------------
// Fused pass: async-LDS B staging + LN stats + logits (WMMA f32 16x16x4)
// + chunk softmax + partial V.  grid (128 chunks, 32 b), block 256 (8 waves).
// ---------------------------------------------------------------------------
__global__ __launch_bounds__(256) void attn_kernel(const float* __restrict__ x,
                                                   const float* __restrict__ gamma,
                                                   const float* __restrict__ beta,
                                                   const float* __restrict__ qk,
                                                   const float* __restrict__ qb,
                                                   float* __restrict__ pm,
                                                   float* __restrict__ ps,
                                                   float* __restrict__ po) {
    __shared__ float xs[CHT][PAD];        // 98816 B staged x tile
    __shared__ float sqk[Cn * NP];        // 49152 B async-staged B matrix
    __shared__ float smu[CHT], srstd[CHT];
    __shared__ float sg[Cn], sb[Cn];
    __shared__ float slog[CHT][NP];
    __shared__ float sw[CHT][NP];

    const int chunk = blockIdx.x, b = blockIdx.y, tid = threadIdx.x;
    const size_t xoff = ((size_t)b * Ln + (size_t)chunk * CHT) * Cn;

    // --- async DMA: qk[b] (48KB) global -> LDS, overlapped with phase A ----
    // 3072 x b128 transfers; LDS byte address = flat addr low 32 bits.
    {
        const float* qkb = qk + (size_t)b * Cn * NP;
        unsigned lds0 = (unsigned)(unsigned long long)(&sqk[0]);
        #pragma unroll
        for (int it = 0; it < 12; ++it) {
            unsigned j = (unsigned)tid + 256u * it;      // 16B chunk index
            unsigned ldsa = lds0 + j * 16u;
            unsigned voff = j * 16u;
            asm volatile("global_load_async_to_lds_b128 %0, %1, %2"
                         :: "v"(ldsa), "v"(voff), "s"(qkb)
                         : "memory");
        }
    }

    for (int i = tid; i < Cn; i += 256) { sg[i] = gamma[i]; sb[i] = beta[i]; }

    // --- phase A: load tile + LN stats (8 threads per token) ---------------
    const int tok = tid >> 3, sub = tid & 7;          // 32 tokens x 8 threads
    const float* xr = x + xoff + (size_t)tok * Cn + sub * 96;
    float s = 0.f, ss = 0.f;
    for (int i = 0; i < 96; i += 4) {
        float4 v = *(const float4*)(xr + i);
        *(float4*)&xs[tok][sub * 96 + i] = v;
        s  += v.x + v.y + v.z + v.w;
        ss += v.x * v.x + v.y * v.y + v.z * v.z + v.w * v.w;
    }
    for (int off = 4; off; off >>= 1) {
        s  += __shfl_xor(s,  off, 8);
        ss += __shfl_xor(ss, off, 8);
    }
    if (sub == 0) {
        float mu  = s * (1.0f / Cn);
        float var = ss * (1.0f / Cn) - mu * mu;
        smu[tok]   = mu;
        srstd[tok] = rsqrtf(var + EPS);
    }
    asm volatile("s_wait_asynccnt 0x0" ::: "memory");  // DMA writes landed
    __syncthreads();

    // --- phase B: logits tile [32x16] via v_wmma_f32_16x16x4_f32 -----------
    const int wave = tid >> 5, lane = tid & 31;
    if (wave < 2) {                                  // EXEC all-ones per wave
        v8f acc = {};
        const int mbase = wave * 16;
        const int row   = mbase + (lane & 15);       // A: M = lane (per half)
        const int koff  = (lane < 16) ? 0 : 2;       // A halves: K+0/1 vs K+2/3
        const int n     = lane & 15;                 // B/D: N across lanes
        const float mu = smu[row], rs = srstd[row];
        for (int k = 0; k < Cn; k += 4) {
            const int ka = k + koff;
            v2f A, Bf;
            A.x = (xs[row][ka]     - mu) * rs * sg[ka]     + sb[ka];
            A.y = (xs[row][ka + 1] - mu) * rs * sg[ka + 1] + sb[ka + 1];
            Bf.x = sqk[ka * NP + n];
            Bf.y = sqk[(ka + 1) * NP + n];
            acc = __builtin_amdgcn_wmma_f32_16x16x4_f32(
                false, A, false, Bf, (short)0, acc, false, false);
        }
        const float qbn = qb[b * NP + n];
        #pragma unroll
        for (int i = 0; i < 8; ++i) {                // D: M = i (+8 upper half)
            int m = mbase + i + ((lane < 16) ? 0 : 8);
            slog[m][n] = acc[i] + qbn;
        }
    }
    __syncthreads();

    // --- phase C: chunk-local online softmax + partial attn@V --------------
    if (tid < Hn) {
        const int h = tid;
        float m = -INFINITY;
        for (int l = 0; l < CHT; ++l) m = fmaxf(m, slog[l][h]);
        float sum = 0.f;
        for (int l = 0; l < CHT; ++l) {
            float w = __expf(slog[l][h] - m);
            sw[l][h] = w;
            sum += w;
        }
        pm[((size_t)b * NCH + chunk) * NP + h] = m;
        ps[((size_t)b * NCH + chunk) * NP + h] = sum;
    }
    __syncthreads();

    float* pob = po + ((size_t)b * NCH + chunk) * Cn;
    for (int c = tid; c < Cn; c += 256) {
        const int h = c >> 6;
        const float g = sg[c], be = sb[c];
        float acc = 0.f;
        for (int l = 0; l < CHT; ++l) {
            float kv = (xs[l][c] - smu[l]) * srstd[l] * g + be;
            acc += sw[l][h] * kv;
        }
        pob[c] = acc;
    }
}

// grid (32 b), block 256: combine 128 chunk partials with exp rescale
__global__ __launch_bounds__(256) void reduce_kernel(const float* __restrict__ pm,
                                                     const float* __restrict__ ps,
                                                     const float* __restrict__ po,
                                                     float* __restrict__ out) {
    __shared__ float Mh[NP], Sh[NP];
    __shared__ float sf[NCH][NP];
    const int b = blockIdx.x, tid = threadIdx.x;
    if (tid < Hn) {
        float M = -INFINITY;
        for (int i = 0; i < NCH; ++i)
            M = fmaxf(M, pm[((size_t)b * NCH + i) * NP + tid]);
        float S = 0.f;
        for (int i = 0; i < NCH; ++i)
            S += ps[((size_t)b * NCH + i) * NP + tid] *
                 __expf(pm[((size_t)b * NCH + i) * NP + tid] - M);
        Mh[tid] = M; Sh[tid] = S;
    }
    __syncthreads();
    for (int i = tid; i < NCH * Hn; i += 256) {
        int ch = i / Hn, h = i % Hn;
        sf[ch][h] = __expf(pm[((size_t)b * NCH + ch) * NP + h] - Mh[h]);
    }
    __syncthreads();
    for (int c = tid; c < Cn; c += 256) {
        const int h = c >> 6;
        float acc = 0.f;
        for (int i = 0; i < NCH; ++i)
            acc += po[((size_t)b * NCH + i) * Cn + c] * sf[i][h];
        out[b * Cn + c] = acc / Sh[h];
    }
}

// ---------------------------------------------------------------------------
extern "C" void kernel_launch(void* const* d_in, const int* in_sizes, int n_in,
                              void* d_out, int out_size, void* d_ws, size_t ws_size,
                              hipStream_t stream) {
    const float* x     = (const float*)d_in[0];
    const float* gamma = (const float*)d_in[1];
    const float* beta  = (const float*)d_in[2];
    const float* Wq    = (const float*)d_in[3];
    const float* bq    = (const float*)d_in[4];
    const float* Wk    = (const float*)d_in[5];
    const float* bk    = (const float*)d_in[6];
    float* out = (float*)d_out;

    float* ws   = (float*)d_ws;
    float* pool = ws + WS_POOL;
    float* qs   = ws + WS_QS;
    float* qk   = ws + WS_QK;
    float* qb   = ws + WS_QB;
    float* pm   = ws + WS_PM;
    float* ps   = ws + WS_PS;
    float* po   = ws + WS_PO;

    zero_pool<<<dim3((Bn * Cn + 255) / 256), dim3(256), 0, stream>>>(pool);
    pool_kernel<<<dim3(32, Bn), dim3(192), 0, stream>>>(x, pool);
    qproj_kernel<<<dim3(Bn), dim3(256), 0, stream>>>(pool, Wq, bq, qs);
    qkbuild_kernel<<<dim3(NP, Bn), dim3(256), 0, stream>>>(qs, Wk, bk, qk, qb);
    attn_kernel<<<dim3(NCH, Bn), dim3(256), 0, stream>>>(x, gamma, beta, qk, qb,
                                                         pm, ps, po);
    reduce_kernel<<<dim3(Bn), dim3(256), 0, stream>>>(pm, ps, po, out);
}